// GraphMultiHeadAttentionLayer_31387620999375
// MI455X (gfx1250) — compile-verified
//
#include <hip/hip_runtime.h>

#define N_NODES 50000
#define DIM     128
#define NEDGE   500000
#define NHEAD   4
#define FDIM    32

// padded row stride (floats) for LDS tiles: kills stride-128 bank conflicts
#define PSTRIDE 130
// LDS layout (floats): Wt1 | Wt2 | A | H   (4 x 128*130 = 266,240 bytes)
#define OFF_WT1 0
#define OFF_WT2 (128 * PSTRIDE)
#define OFF_A   (2 * 128 * PSTRIDE)
#define OFF_H   (3 * 128 * PSTRIDE)

typedef __attribute__((ext_vector_type(2))) float    v2f;
typedef __attribute__((ext_vector_type(8))) float    v8f;
typedef __attribute__((ext_vector_type(4))) unsigned v4u;
typedef __attribute__((ext_vector_type(8))) int      v8i;
typedef __attribute__((ext_vector_type(4))) int      v4i;

// ---- monotone float <-> uint mapping for atomicMax on floats ----
__device__ __forceinline__ unsigned f2mono(float f) {
    unsigned u = __float_as_uint(f);
    return (u & 0x80000000u) ? ~u : (u | 0x80000000u);
}
__device__ __forceinline__ float mono2f(unsigned u) {
    unsigned b = (u & 0x80000000u) ? (u & 0x7FFFFFFFu) : ~u;
    return __uint_as_float(b);
}

// =====================================================================
// Fused 2-layer MLP: q = relu(LN2(relu(LN1(nodes@W1+b1)) @ W2 + b2))
// 8 waves / block, each wave owns a 16-row tile (block = 128 rows).
// A tile pulled into LDS by the Tensor Data Mover (2D tile, HW padding
// to a 130-float row stride); W1/W2 transposed into LDS by the waves in
// parallel with the TDM transfer. GEMMs via V_WMMA_F32_16X16X4_F32.
// =====================================================================
__global__ __launch_bounds__(256) void mlp_wmma_kernel(
    const float* __restrict__ nodes,
    const float* __restrict__ W1, const float* __restrict__ b1,
    const float* __restrict__ ln1s, const float* __restrict__ ln1b,
    const float* __restrict__ W2, const float* __restrict__ b2,
    const float* __restrict__ ln2s, const float* __restrict__ ln2b,
    float* __restrict__ q)
{
    extern __shared__ float smem[];
    float* sWt1 = smem + OFF_WT1;   // W1^T, [col][k], stride PSTRIDE
    float* sWt2 = smem + OFF_WT2;   // W2^T
    float* sA   = smem + OFF_A;     // 128 x 128 node tile, stride PSTRIDE (TDM)
    float* sH   = smem + OFF_H;     // intermediate h tile, stride PSTRIDE

    const int tid  = threadIdx.x;
    const int wave = tid >> 5;

    // ---- issue TDM load of the A tile (wave 0 only; EXEC-independent op) ----
    // D# built per CDNA5 ISA.
    // 2D tile: tile 128x128 f32 from nodes[blockIdx*128 ..], tensor dims
    // (128, N_NODES) -> OOB tail rows are zero-filled by TDM.
    // pad_enable: +2 DWORDs every 128 DWORDs => LDS row stride 130 floats.
    if (wave == 0) {
        unsigned long long ga =
            (unsigned long long)(nodes + (size_t)blockIdx.x * 128 * DIM);
        v4u g0;
        g0[0] = 1u;                                   // count=1 (valid), user desc
        g0[1] = (unsigned)(OFF_A * sizeof(float));    // lds_addr (dyn LDS base = 0)
        g0[2] = (unsigned)(ga & 0xFFFFFFFFull);       // global_addr[31:0]
        g0[3] = (unsigned)((ga >> 32) & 0x01FFFFFFull) | (2u << 30); // addr[56:32] | type=2
        v8i g1;
        g1[0] = (int)((2u << 16)      // data_size = 4 bytes
                    | (1u << 20)      // pad_enable
                    | (6u << 22)      // pad_interval: 128 DWORDs
                    | (1u << 25));    // pad_amount: 2 DWORDs
        g1[1] = (int)(128u << 16);                    // tensor_dim0 = 128
        g1[2] = (int)((unsigned)N_NODES << 16);       // tensor_dim1[15:0]
        g1[3] = (int)(((unsigned)N_NODES >> 16) | (128u << 16)); // dim1[31:16] | tile_dim0
        g1[4] = 128;                                  // tile_dim1 = 128 (tile_dim2 = 0)
        g1[5] = 128;                                  // tensor_dim0_stride = 128
        g1[6] = 0;
        g1[7] = 0;
        v4i gz4 = {0, 0, 0, 0};                       // 2D tensor: groups 2/3 unused
        v8i gz8 = {0, 0, 0, 0, 0, 0, 0, 0};
        __builtin_amdgcn_tensor_load_to_lds(g0, g1, gz4, gz4, gz8, 0);
    }

    // ---- transpose-stage W1/W2 into LDS (overlaps with TDM transfer) ----
    // read W[k][c] coalesced, write Wt[c][k]; per-wave bank-conflict free.
    for (int i = tid; i < DIM * DIM; i += 256) {
        int k = i >> 7, c = i & 127;
        sWt1[c * PSTRIDE + k] = W1[i];
        sWt2[c * PSTRIDE + k] = W2[i];
    }

    if (wave == 0) __builtin_amdgcn_s_wait_tensorcnt(0);
    __syncthreads();

    const int lane = tid & 31;
    const int half = lane >> 4;     // 0 | 1
    const int l    = lane & 15;     // N index / A row within tile
    const int rowbase = blockIdx.x * 128 + wave * 16;

    v8f acc[8];

    // ---------------- layer 1: A(sA) @ W1(sWt1) ----------------
    #pragma unroll
    for (int nt = 0; nt < 8; ++nt)
        #pragma unroll
        for (int j = 0; j < 8; ++j) acc[nt][j] = 0.0f;

    {
        const float* Arow = sA + (wave * 16 + l) * PSTRIDE;
        for (int k = 0; k < DIM; k += 4) {
            v2f a = *(const v2f*)(Arow + k + 2 * half);
            #pragma unroll
            for (int nt = 0; nt < 8; ++nt) {
                v2f b = *(const v2f*)(sWt1 + (nt * 16 + l) * PSTRIDE + k + 2 * half);
                acc[nt] = __builtin_amdgcn_wmma_f32_16x16x4_f32(
                    false, a, false, b, (short)0, acc[nt], false, false);
            }
        }
    }

    // bias + LayerNorm1 + relu -> sH
    // C layout: element (lane, vgpr j) = row (j + 8*half), col (nt*16 + l)
    {
        #pragma unroll
        for (int nt = 0; nt < 8; ++nt) {
            float bv = b1[nt * 16 + l];
            #pragma unroll
            for (int j = 0; j < 8; ++j) acc[nt][j] += bv;
        }
        float s1[8], s2[8];
        #pragma unroll
        for (int j = 0; j < 8; ++j) { s1[j] = 0.f; s2[j] = 0.f; }
        #pragma unroll
        for (int nt = 0; nt < 8; ++nt)
            #pragma unroll
            for (int j = 0; j < 8; ++j) { float v = acc[nt][j]; s1[j] += v; s2[j] += v * v; }
        #pragma unroll
        for (int m = 1; m <= 8; m <<= 1)
            #pragma unroll
            for (int j = 0; j < 8; ++j) {
                s1[j] += __shfl_xor(s1[j], m, 32);
                s2[j] += __shfl_xor(s2[j], m, 32);
            }
        #pragma unroll
        for (int nt = 0; nt < 8; ++nt) {
            int col = nt * 16 + l;
            float sc = ln1s[col], bi = ln1b[col];
            #pragma unroll
            for (int j = 0; j < 8; ++j) {
                float mean = s1[j] * (1.0f / 128.0f);
                float var  = s2[j] * (1.0f / 128.0f) - mean * mean;
                float v = (acc[nt][j] - mean) * rsqrtf(var + 1e-6f) * sc + bi;
                sH[(wave * 16 + j + 8 * half) * PSTRIDE + col] = fmaxf(v, 0.0f);
            }
        }
    }
    __syncthreads();

    // ---------------- layer 2: h(sH) @ W2(sWt2) ----------------
    #pragma unroll
    for (int nt = 0; nt < 8; ++nt)
        #pragma unroll
        for (int j = 0; j < 8; ++j) acc[nt][j] = 0.0f;

    {
        const float* Hrow = sH + (wave * 16 + l) * PSTRIDE;
        for (int k = 0; k < DIM; k += 4) {
            v2f a = *(const v2f*)(Hrow + k + 2 * half);
            #pragma unroll
            for (int nt = 0; nt < 8; ++nt) {
                v2f b = *(const v2f*)(sWt2 + (nt * 16 + l) * PSTRIDE + k + 2 * half);
                acc[nt] = __builtin_amdgcn_wmma_f32_16x16x4_f32(
                    false, a, false, b, (short)0, acc[nt], false, false);
            }
        }
    }

    // bias + LayerNorm2 + relu -> q (global, guarded tail)
    {
        #pragma unroll
        for (int nt = 0; nt < 8; ++nt) {
            float bv = b2[nt * 16 + l];
            #pragma unroll
            for (int j = 0; j < 8; ++j) acc[nt][j] += bv;
        }
        float s1[8], s2[8];
        #pragma unroll
        for (int j = 0; j < 8; ++j) { s1[j] = 0.f; s2[j] = 0.f; }
        #pragma unroll
        for (int nt = 0; nt < 8; ++nt)
            #pragma unroll
            for (int j = 0; j < 8; ++j) { float v = acc[nt][j]; s1[j] += v; s2[j] += v * v; }
        #pragma unroll
        for (int m = 1; m <= 8; m <<= 1)
            #pragma unroll
            for (int j = 0; j < 8; ++j) {
                s1[j] += __shfl_xor(s1[j], m, 32);
                s2[j] += __shfl_xor(s2[j], m, 32);
            }
        #pragma unroll
        for (int nt = 0; nt < 8; ++nt) {
            int col = nt * 16 + l;
            float sc = ln2s[col], bi = ln2b[col];
            #pragma unroll
            for (int j = 0; j < 8; ++j) {
                int r = rowbase + j + 8 * half;
                if (r < N_NODES) {
                    float mean = s1[j] * (1.0f / 128.0f);
                    float var  = s2[j] * (1.0f / 128.0f) - mean * mean;
                    float v = (acc[nt][j] - mean) * rsqrtf(var + 1e-6f) * sc + bi;
                    q[(size_t)r * DIM + col] = fmaxf(v, 0.0f);
                }
            }
        }
    }
}

// =====================================================================
// init: zero max/denom/agg buffers (d_ws is poisoned, not re-zeroed)
// =====================================================================
__global__ void init_kernel(unsigned* __restrict__ maxu,
                            float* __restrict__ denom,
                            float* __restrict__ agg)
{
    int i = blockIdx.x * blockDim.x + threadIdx.x;
    if (i < N_NODES * NHEAD) { maxu[i] = 0u; denom[i] = 0.0f; }
    if (i < N_NODES * DIM) agg[i] = 0.0f;
}

// =====================================================================
// warp-per-edge: per-head logits + segment max (monotone-uint atomicMax)
// lane owns float4 of the 128 features; 8-lane groups = one head
// =====================================================================
__global__ __launch_bounds__(256) void edge_logits_kernel(
    const float* __restrict__ q,
    const int* __restrict__ senders, const int* __restrict__ receivers,
    float* __restrict__ logits, unsigned* __restrict__ maxu)
{
    int e = (blockIdx.x * blockDim.x + threadIdx.x) >> 5;
    int lane = threadIdx.x & 31;
    if (e >= NEDGE) return;
    int s = senders[e], r = receivers[e];
    float4 sv = ((const float4*)(q + (size_t)s * DIM))[lane];
    float4 rv = ((const float4*)(q + (size_t)r * DIM))[lane];
    float p = sv.x * rv.x + sv.y * rv.y + sv.z * rv.z + sv.w * rv.w;
    p += __shfl_xor(p, 1, 32);
    p += __shfl_xor(p, 2, 32);
    p += __shfl_xor(p, 4, 32);
    if ((lane & 7) == 0) {
        int h = lane >> 3;
        float lg = p * 0.17677669529663687f;   // 1/sqrt(32)
        logits[(size_t)e * NHEAD + h] = lg;
        atomicMax(&maxu[(size_t)r * NHEAD + h], f2mono(lg));
    }
}

// =====================================================================
// unnormalized weights + denominator (segment sum, HW f32 atomics)
// =====================================================================
__global__ void edge_denom_kernel(const int* __restrict__ receivers,
                                  float* __restrict__ logits,
                                  const unsigned* __restrict__ maxu,
                                  float* __restrict__ denom)
{
    int i = blockIdx.x * blockDim.x + threadIdx.x;
    if (i >= NEDGE * NHEAD) return;
    int e = i >> 2, h = i & 3;
    int r = receivers[e];
    float m = mono2f(maxu[(size_t)r * NHEAD + h]);
    float w = __expf(logits[i] - m);
    logits[i] = w;                         // rewritten every launch -> deterministic
    unsafeAtomicAdd(&denom[(size_t)r * NHEAD + h], w);
}

// =====================================================================
// warp-per-edge weighted message scatter (native f32 atomics, L2-resident)
// =====================================================================
__global__ __launch_bounds__(256) void edge_scatter_kernel(
    const float* __restrict__ q,
    const int* __restrict__ senders, const int* __restrict__ receivers,
    const float* __restrict__ wun, const float* __restrict__ denom,
    float* __restrict__ agg)
{
    int e = (blockIdx.x * blockDim.x + threadIdx.x) >> 5;
    int lane = threadIdx.x & 31;
    if (e >= NEDGE) return;
    int s = senders[e], r = receivers[e];
    int h = lane >> 3;
    float w = wun[(size_t)e * NHEAD + h] / denom[(size_t)r * NHEAD + h];
    float4 sv = ((const float4*)(q + (size_t)s * DIM))[lane];
    float* dst = agg + (size_t)r * DIM + lane * 4;
    unsafeAtomicAdd(dst + 0, sv.x * w);
    unsafeAtomicAdd(dst + 1, sv.y * w);
    unsafeAtomicAdd(dst + 2, sv.z * w);
    unsafeAtomicAdd(dst + 3, sv.w * w);
}

// =====================================================================
// out[n][f] = mean_h relu(agg[n][h][f])
// =====================================================================
__global__ void finalize_kernel(const float* __restrict__ agg,
                                float* __restrict__ out)
{
    int i = blockIdx.x * blockDim.x + threadIdx.x;
    if (i >= N_NODES * FDIM) return;
    int n = i >> 5, f = i & 31;
    const float* a = agg + (size_t)n * DIM;
    float s = 0.0f;
    #pragma unroll
    for (int h = 0; h < NHEAD; ++h) s += fmaxf(a[h * FDIM + f], 0.0f);
    out[i] = s * 0.25f;
}

extern "C" void kernel_launch(void* const* d_in, const int* in_sizes, int n_in,
                              void* d_out, int out_size, void* d_ws, size_t ws_size,
                              hipStream_t stream)
{
    const float* nodes = (const float*)d_in[0];
    const float* W1    = (const float*)d_in[1];
    const float* b1    = (const float*)d_in[2];
    const float* ln1s  = (const float*)d_in[3];
    const float* ln1b  = (const float*)d_in[4];
    const float* W2    = (const float*)d_in[5];
    const float* b2    = (const float*)d_in[6];
    const float* ln2s  = (const float*)d_in[7];
    const float* ln2b  = (const float*)d_in[8];
    const int* senders   = (const int*)d_in[9];
    const int* receivers = (const int*)d_in[10];
    float* out = (float*)d_out;

    // workspace partition (all fits in 192MB L2: q 25.6MB, logits 8MB, agg 25.6MB)
    float*    q      = (float*)d_ws;                                  // N*128
    float*    logits = q + (size_t)N_NODES * DIM;                     // E*4
    unsigned* maxu   = (unsigned*)(logits + (size_t)NEDGE * NHEAD);   // N*4
    float*    denom  = (float*)(maxu + (size_t)N_NODES * NHEAD);      // N*4
    float*    agg    = denom + (size_t)N_NODES * NHEAD;               // N*128

    {
        int total = N_NODES * DIM;
        init_kernel<<<(total + 255) / 256, 256, 0, stream>>>(maxu, denom, agg);
    }
    {
        int grid = (N_NODES + 127) / 128;
        size_t shmem = (size_t)(4 * 128 * PSTRIDE) * sizeof(float); // 266,240 B
        mlp_wmma_kernel<<<grid, 256, shmem, stream>>>(
            nodes, W1, b1, ln1s, ln1b, W2, b2, ln2s, ln2b, q);
    }
    edge_logits_kernel<<<(NEDGE * 32 + 255) / 256, 256, 0, stream>>>(
        q, senders, receivers, logits, maxu);
    edge_denom_kernel<<<(NEDGE * NHEAD + 255) / 256, 256, 0, stream>>>(
        receivers, logits, maxu, denom);
    edge_scatter_kernel<<<(NEDGE * 32 + 255) / 256, 256, 0, stream>>>(
        q, senders, receivers, logits, denom, agg);
    finalize_kernel<<<(N_NODES * FDIM + 255) / 256, 256, 0, stream>>>(agg, out);
}